// MotionCorrect_60318520705514
// MI455X (gfx1250) — compile-verified
//
#include <hip/hip_runtime.h>
#include <math.h>

typedef float v2f __attribute__((ext_vector_type(2)));
typedef float v8f __attribute__((ext_vector_type(8)));

#define PI_F 3.14159265358979323846f
#define NPix 262144      // 512*512
#define NB   32

__device__ __forceinline__ v8f wmma4(v2f a, v2f b, v8f c) {
    // D = A(16x4) * B(4x16) + C : V_WMMA_F32_16X16X4_F32
    return __builtin_amdgcn_wmma_f32_16x16x4_f32(false, a, false, b, (short)0, c, false, false);
}

// ---------------------------------------------------------------------------
// 512-point complex FFT per wave via four-step 512 = 16 x 32, all matmuls on
// the WMMA pipe.  n = 32*n1 + n2,  k = k1 + 16*k2.
//   step1: Y[k1][n2] = sum_n1 x[32 n1+n2] * W16^(n1 k1)       (D16 * M)
//   step2: Z = Y * W512^(k1 n2)
//   step3: X[k1][k2] = sum_n2 Z[k1][n2] * W32^(n2 k2)         (Z * D32)
// Input row in per-wave LDS re/im[512] (destroyed). Result in C-layout regs.
// ---------------------------------------------------------------------------
template<bool INV>
__device__ void fft512_core(float* re, float* im,
                            v8f& xr0, v8f& xi0, v8f& xr1, v8f& xi1)
{
    const int lane = threadIdx.x & 31;
    const int hi   = lane >> 4;
    const int m    = lane & 15;
    const float sgn = INV ? 1.0f : -1.0f;

    // A-fragments of the 16-pt DFT matrix, layout: lanes0-15 M=lane,V0:K=b+0,
    // V1:K=b+1; lanes16-31 M=lane-16,V0:K=b+2,V1:K=b+3  =>  K = 4kk+2hi+v.
    v2f d16r[4], d16i[4], d16in[4];
#pragma unroll
    for (int kk = 0; kk < 4; ++kk) {
        int K0 = 4 * kk + 2 * hi;
        float s0, c0, s1, c1;
        __sincosf(2.0f * PI_F * (float)(m * K0)       / 16.0f, &s0, &c0);
        __sincosf(2.0f * PI_F * (float)(m * (K0 + 1)) / 16.0f, &s1, &c1);
        d16r[kk]  = (v2f){ c0, c1 };
        d16i[kk]  = (v2f){ sgn * s0,  sgn * s1 };
        d16in[kk] = (v2f){ -sgn * s0, -sgn * s1 };
    }

    v8f zr[2], zi[2];
#pragma unroll
    for (int t = 0; t < 2; ++t) {
        v8f yr = {0.f,0.f,0.f,0.f,0.f,0.f,0.f,0.f};
        v8f yi = {0.f,0.f,0.f,0.f,0.f,0.f,0.f,0.f};
        const int col = 16 * t + m;
#pragma unroll
        for (int kk = 0; kk < 4; ++kk) {
            int K0 = 4 * kk + 2 * hi;
            v2f br = { re[32 * K0 + col], re[32 * (K0 + 1) + col] };
            v2f bi = { im[32 * K0 + col], im[32 * (K0 + 1) + col] };
            yr = wmma4(d16r[kk],  br, yr);
            yr = wmma4(d16in[kk], bi, yr);   // - Di * Mi
            yi = wmma4(d16r[kk],  bi, yi);
            yi = wmma4(d16i[kk],  br, yi);
        }
        // twiddle W512^(k1*n2); C-layout: (M=r+8hi, N=16t+m)
#pragma unroll
        for (int r = 0; r < 8; ++r) {
            int k1 = r + 8 * hi;
            float s, c;
            __sincosf(sgn * 2.0f * PI_F * (float)(k1 * col) / 512.0f, &s, &c);
            float a = yr[r], b = yi[r];
            zr[t][r] = a * c - b * s;
            zi[t][r] = a * s + b * c;
        }
    }
    // scatter Z back to LDS at [k1*32 + n2] (all M reads finished; same wave,
    // DS instructions retire in order -> no barrier needed)
#pragma unroll
    for (int t = 0; t < 2; ++t)
#pragma unroll
        for (int r = 0; r < 8; ++r) {
            int idx = (r + 8 * hi) * 32 + 16 * t + m;
            re[idx] = zr[t][r];
            im[idx] = zi[t][r];
        }

    // step 3: X = Z (16x32, A operand from LDS) * D32 (32x32, generated)
#pragma unroll
    for (int t = 0; t < 2; ++t) {
        v8f xr = {0.f,0.f,0.f,0.f,0.f,0.f,0.f,0.f};
        v8f xi = {0.f,0.f,0.f,0.f,0.f,0.f,0.f,0.f};
        const int n = 16 * t + m;                 // k2 column
#pragma unroll
        for (int kk = 0; kk < 8; ++kk) {
            int K0 = 4 * kk + 2 * hi;
            v2f ar  = { re[m * 32 + K0], re[m * 32 + K0 + 1] };
            v2f ai  = { im[m * 32 + K0], im[m * 32 + K0 + 1] };
            v2f ain = { -ai[0], -ai[1] };
            float s0, c0, s1, c1;
            __sincosf(sgn * 2.0f * PI_F * (float)(K0 * n)       / 32.0f, &s0, &c0);
            __sincosf(sgn * 2.0f * PI_F * (float)((K0 + 1) * n) / 32.0f, &s1, &c1);
            v2f br = { c0, c1 };
            v2f bi = { s0, s1 };
            xr = wmma4(ar,  br, xr);
            xr = wmma4(ain, bi, xr);
            xi = wmma4(ar,  bi, xi);
            xi = wmma4(ai,  br, xi);
        }
        if (t == 0) { xr0 = xr; xi0 = xi; } else { xr1 = xr; xi1 = xi; }
    }
}

template<bool INV>
__device__ __forceinline__ void fft_write(float2* dst, int stride,
                                          const v8f& xr0, const v8f& xi0,
                                          const v8f& xr1, const v8f& xi1)
{
    const int lane = threadIdx.x & 31, hi = lane >> 4, m = lane & 15;
    const float sc = INV ? (1.0f / 512.0f) : 1.0f;
#pragma unroll
    for (int r = 0; r < 8; ++r) {
        int k1 = r + 8 * hi;
        int ka = k1 + 16 * m;              // t = 0
        int kb = k1 + 16 * (16 + m);       // t = 1
        dst[(size_t)ka * stride] = make_float2(xr0[r] * sc, xi0[r] * sc);
        dst[(size_t)kb * stride] = make_float2(xr1[r] * sc, xi1[r] * sc);
    }
}

__device__ __forceinline__ void load_row_real(const float* src, float mean,
                                              float* re, float* im)
{
    const int lane = threadIdx.x & 31;
#pragma unroll
    for (int j = 0; j < 16; ++j) {
        int n = lane + 32 * j;
        re[n] = src[n] - mean;
        im[n] = 0.0f;
    }
}

__device__ __forceinline__ void load_row_cplx(const float2* src, int stride,
                                              float* re, float* im)
{
    const int lane = threadIdx.x & 31;
#pragma unroll
    for (int j = 0; j < 16; ++j) {
        int n = lane + 32 * j;
        float2 v = src[(size_t)n * stride];
        re[n] = v.x;
        im[n] = v.y;
    }
}

// ------------------------- kernels ----------------------------------------

// per-frame means (blocks 0..31) and template mean / variance (block 32)
__global__ void k_stats(const float* __restrict__ fr, const float* __restrict__ tmpl,
                        float* __restrict__ stats)
{
    __shared__ float r1[256], r2[256];
    int b = blockIdx.x, tid = threadIdx.x;
    const float* s = (b < NB) ? fr + (size_t)b * NPix : tmpl;
    float a = 0.f, a2 = 0.f;
    for (int i = tid; i < NPix; i += 256) { float v = s[i]; a += v; a2 += v * v; }
    r1[tid] = a; r2[tid] = a2; __syncthreads();
    for (int st = 128; st > 0; st >>= 1) {
        if (tid < st) { r1[tid] += r1[tid + st]; r2[tid] += r2[tid + st]; }
        __syncthreads();
    }
    if (tid == 0) {
        float mean = r1[0] * (1.0f / (float)NPix);
        if (b < NB) stats[b] = mean;
        else { stats[32] = mean; stats[33] = r2[0] - (float)NPix * mean * mean + 1e-8f; }
    }
}

// forward FFT of real rows; optional per-chunk mean subtraction (chunk = 512 rows)
__global__ void k_fft_rows_real(const float* __restrict__ src, float2* __restrict__ dst,
                                const float* __restrict__ meanArr, int subMean)
{
    __shared__ float sre[4][512], sim[4][512];
    int wv = blockIdx.x * 4 + (threadIdx.x >> 5);
    float* re = sre[threadIdx.x >> 5];
    float* im = sim[threadIdx.x >> 5];
    float mean = subMean ? meanArr[wv >> 9] : 0.0f;
    load_row_real(src + (size_t)wv * 512, mean, re, im);
    v8f xr0, xi0, xr1, xi1;
    fft512_core<false>(re, im, xr0, xi0, xr1, xi1);
    fft_write<false>(dst + (size_t)wv * 512, 1, xr0, xi0, xr1, xi1);
}

// FFT along columns, complex in-place ((b<<9)|c waves)
template<bool INV>
__global__ void k_fft_cols(float2* data)
{
    __shared__ float sre[4][512], sim[4][512];
    int wv = blockIdx.x * 4 + (threadIdx.x >> 5);
    int b = wv >> 9, c = wv & 511;
    float* re = sre[threadIdx.x >> 5];
    float* im = sim[threadIdx.x >> 5];
    float2* p = data + (size_t)b * NPix + c;
    load_row_cplx(p, 512, re, im);
    v8f xr0, xi0, xr1, xi1;
    fft512_core<INV>(re, im, xr0, xi0, xr1, xi1);
    fft_write<INV>(p, 512, xr0, xi0, xr1, xi1);
}

// FFT along rows, complex in-place
template<bool INV>
__global__ void k_fft_rows_cplx(float2* data)
{
    __shared__ float sre[4][512], sim[4][512];
    int wv = blockIdx.x * 4 + (threadIdx.x >> 5);
    float* re = sre[threadIdx.x >> 5];
    float* im = sim[threadIdx.x >> 5];
    float2* p = data + (size_t)wv * 512;
    load_row_cplx(p, 1, re, im);
    v8f xr0, xi0, xr1, xi1;
    fft512_core<INV>(re, im, xr0, xi0, xr1, xi1);
    fft_write<INV>(p, 1, xr0, xi0, xr1, xi1);
}

// F *= conj(T)
__global__ void k_product(float2* __restrict__ F, const float2* __restrict__ T)
{
    size_t i = (size_t)blockIdx.x * 256 + threadIdx.x;
    int pix = (int)(i & (NPix - 1));
    float2 a = F[i], t = T[pix];
    F[i] = make_float2(a.x * t.x + a.y * t.y, a.y * t.x - a.x * t.y);
}

// inverse row FFT only for the 11 rolled rows; emit |cc| into 11x11 window
__global__ void k_fft_rows_ncc(const float2* __restrict__ F, float* __restrict__ nom)
{
    __shared__ float sre[4][512], sim[4][512];
    int wv = blockIdx.x * 4 + (threadIdx.x >> 5);   // 0 .. 32*11-1
    int b = wv / 11, p = wv % 11;
    int row = (507 + p) & 511;                      // rolled row -> cc row
    float* re = sre[threadIdx.x >> 5];
    float* im = sim[threadIdx.x >> 5];
    load_row_cplx(F + (size_t)b * NPix + (size_t)row * 512, 1, re, im);
    v8f xr0, xi0, xr1, xi1;
    fft512_core<true>(re, im, xr0, xi0, xr1, xi1);
    const int lane = threadIdx.x & 31, hi = lane >> 4, m = lane & 15;
    const float sc = 1.0f / 512.0f;
#pragma unroll
    for (int t = 0; t < 2; ++t)
#pragma unroll
        for (int r = 0; r < 8; ++r) {
            int k = (r + 8 * hi) + 16 * (16 * t + m);
            int q = (k <= 5) ? k + 5 : ((k >= 507) ? k - 507 : -1);
            if (q >= 0) {
                float a = (t ? xr1[r] : xr0[r]) * sc;
                float c = (t ? xi1[r] : xi0[r]) * sc;
                nom[b * 121 + p * 11 + q] = sqrtf(a * a + c * c);
            }
        }
}

// local window sums by inclusion-exclusion -> denominator (one block / frame)
__global__ void k_window(const float* __restrict__ fr, const float* __restrict__ tstats,
                         float* __restrict__ denom)
{
    __shared__ float r1[256], r2[256];
    __shared__ float rowS[20][2], colS[20][2];
    __shared__ float patch[400][2];                 // 4 x 10x10 corners, (x, x^2)
    int b = blockIdx.x, tid = threadIdx.x;
    if (tid < 20) { rowS[tid][0] = rowS[tid][1] = 0.f; colS[tid][0] = colS[tid][1] = 0.f; }
    __syncthreads();
    const float* s = fr + (size_t)b * NPix;
    float a = 0.f, a2 = 0.f;
    for (int i = tid; i < NPix; i += 256) {
        float v = s[i], v2 = v * v;
        a += v; a2 += v2;
        int rr = i >> 9, cc = i & 511;
        int re = (rr < 10) ? rr : ((rr >= 502) ? rr - 492 : -1);
        if (re >= 0) { atomicAdd(&rowS[re][0], v); atomicAdd(&rowS[re][1], v2); }
        int ce = (cc < 10) ? cc : ((cc >= 502) ? cc - 492 : -1);
        if (ce >= 0) { atomicAdd(&colS[ce][0], v); atomicAdd(&colS[ce][1], v2); }
    }
    r1[tid] = a; r2[tid] = a2;
    for (int t2 = tid; t2 < 400; t2 += 256) {
        int pid = t2 / 100, li = (t2 % 100) / 10, lj = t2 % 10;
        int pi = (pid < 2) ? li : 502 + li;
        int pj = (pid & 1) ? 502 + lj : lj;
        float v = s[(size_t)pi * 512 + pj];
        patch[t2][0] = v; patch[t2][1] = v * v;
    }
    __syncthreads();
    for (int st = 128; st > 0; st >>= 1) {
        if (tid < st) { r1[tid] += r1[tid + st]; r2[tid] += r2[tid + st]; }
        __syncthreads();
    }
    if (tid < 121) {
        int p = tid / 11, q = tid % 11;
        const float inv_k = 1.0f / (502.0f * 502.0f);
        float w[2];
        for (int cmp = 0; cmp < 2; ++cmp) {
            float S = cmp ? r2[0] : r1[0];
            float T = 0.f, B = 0.f, L = 0.f, R = 0.f;
            for (int e = 0; e < p; ++e)      T += rowS[e][cmp];
            for (int e = p + 10; e < 20; ++e) B += rowS[e][cmp];
            for (int e = 0; e < q; ++e)      L += colS[e][cmp];
            for (int e = q + 10; e < 20; ++e) R += colS[e][cmp];
            float C = 0.f;
            for (int li = 0; li < 10; ++li)
                for (int lj = 0; lj < 10; ++lj) {
                    int o = li * 10 + lj;
                    if (li < p  && lj < q ) C += patch[0 * 100 + o][cmp];   // TL
                    if (li < p  && lj >= q) C += patch[1 * 100 + o][cmp];   // TR
                    if (li >= p && lj < q ) C += patch[2 * 100 + o][cmp];   // BL
                    if (li >= p && lj >= q) C += patch[3 * 100 + o][cmp];   // BR
                }
            w[cmp] = S - T - B - L - R + C;
        }
        float ls  = w[0] * inv_k;
        float ls2 = w[1] * inv_k;
        float var = ls2 - ls * ls * inv_k + 1e-8f;
        if (var < 0.f) var = 0.f;
        denom[b * 121 + tid] = sqrtf(tstats[1] * var);
    }
}

// NCC, argmax, log-parabola sub-pixel shifts (one block / frame)
__global__ void k_shift(const float* __restrict__ nom, const float* __restrict__ den,
                        float* __restrict__ shifts)
{
    __shared__ float ncc[121];
    __shared__ float bv[128];
    __shared__ int   bi[128];
    int b = blockIdx.x, tid = threadIdx.x;
    if (tid < 121) {
        float v = nom[b * 121 + tid] / den[b * 121 + tid];
        if (v != v) v = 0.0f;
        ncc[tid] = v;
    }
    __syncthreads();
    float best = -3.4e38f; int besti = 0;
    for (int i = tid; i < 121; i += 128)
        if (ncc[i] > best) { best = ncc[i]; besti = i; }
    bv[tid] = best; bi[tid] = besti; __syncthreads();
    for (int st = 64; st > 0; st >>= 1) {
        if (tid < st) {
            if (bv[tid + st] > bv[tid] ||
                (bv[tid + st] == bv[tid] && bi[tid + st] < bi[tid])) {
                bv[tid] = bv[tid + st]; bi[tid] = bi[tid + st];
            }
        }
        __syncthreads();
    }
    if (tid == 0) {
        int am = bi[0], sx = am / 11, sy = am % 11;
        float shx = -(float)(sx - 5), shy = -(float)(sy - 5);
        auto L = [&](int i, int j) { return logf(ncc[i * 11 + j]); };
        int im1 = (sx - 1 < 0) ? 10 : sx - 1;     // JAX negative-index wrap
        int ip1 = (sx + 1 > 10) ? 10 : sx + 1;    // JAX OOB clamp
        int jm1 = (sy - 1 < 0) ? 10 : sy - 1;
        int jp1 = (sy + 1 > 10) ? 10 : sy + 1;
        float lxm = L(im1, sy), lxp = L(ip1, sy);
        float lym = L(sx, jm1), lyp = L(sx, jp1);
        float l4 = 4.0f * L(sx, sy);
        shx -= (lxm - lxp) / (2.f * lxm - l4 + 2.f * lxp);
        shy -= (lym - lyp) / (2.f * lym - l4 + 2.f * lyp);
        shifts[2 * b] = shx; shifts[2 * b + 1] = shy;
    }
}

// Greg = src_freq * exp(-2*pi*i*(sh_y*Nr/W + sh_x*Nc/H))
__global__ void k_phase(float2* __restrict__ F, const float* __restrict__ shifts)
{
    size_t idx = (size_t)blockIdx.x * 256 + threadIdx.x;
    int b = (int)(idx >> 18);
    int pix = (int)(idx & (NPix - 1));
    int i = pix >> 9, j = pix & 511;
    float Nr = (j < 256) ? (float)j : (float)(j - 512);
    float Nc = (i < 256) ? (float)i : (float)(i - 512);
    float phi = 2.0f * PI_F * (shifts[2 * b + 1] * Nr + shifts[2 * b] * Nc) / 512.0f;
    float s, c;
    __sincosf(phi, &s, &c);
    float2 a = F[idx];
    F[idx] = make_float2(a.x * c + a.y * s, a.y * c - a.x * s);   // a * e^{-i phi}
}

// out[b, j*H + i] = Re(new_img[b, i, j])   (transpose(0,2,1).reshape)
__global__ void k_writeout(const float2* __restrict__ F, float* __restrict__ out)
{
    size_t idx = (size_t)blockIdx.x * 256 + threadIdx.x;
    int b = (int)(idx >> 18);
    int pix = (int)(idx & (NPix - 1));
    int i = pix >> 9, j = pix & 511;
    out[(size_t)b * NPix + (size_t)j * 512 + i] = F[idx].x;
}

// ---------------------------------------------------------------------------
extern "C" void kernel_launch(void* const* d_in, const int* in_sizes, int n_in,
                              void* d_out, int out_size, void* d_ws, size_t ws_size,
                              hipStream_t stream)
{
    const float* fr   = (const float*)d_in[0];   // (1,32,512,512,1) f32
    const float* tmpl = (const float*)d_in[1];   // (512,512) f32
    float* out = (float*)d_out;

    float2* F      = (float2*)d_ws;                          // 32*512*512 complex
    float2* T      = F + (size_t)NB * NPix;                  // 512*512 complex
    float*  stats  = (float*)(T + NPix);                     // [0..31] means, [32] tmean, [33] tvar
    float*  nom    = stats + 64;                             // 32*121
    float*  den    = nom + NB * 121;                         // 32*121
    float*  shifts = den + NB * 121;                         // 32*2

    // 1) statistics
    k_stats<<<NB + 1, 256, 0, stream>>>(fr, tmpl, stats);

    // 2) template FFT2 (zero-mean) -> T
    k_fft_rows_real<<<128, 128, 0, stream>>>(tmpl, T, stats + 32, 1);
    k_fft_cols<false><<<128, 128, 0, stream>>>(T);

    // 3) zero-mean frame FFT2 -> F  (batch-axis FFT cancels analytically)
    k_fft_rows_real<<<4096, 128, 0, stream>>>(fr, F, stats, 1);
    k_fft_cols<false><<<4096, 128, 0, stream>>>(F);

    // 4) cross-power spectrum, inverse transform, 11x11 |cc| window
    k_product<<<32768, 256, 0, stream>>>(F, T);
    k_fft_cols<true><<<4096, 128, 0, stream>>>(F);
    k_fft_rows_ncc<<<88, 128, 0, stream>>>(F, nom);

    // 5) NCC denominator + sub-pixel shifts
    k_window<<<NB, 256, 0, stream>>>(fr, stats + 32, den);
    k_shift<<<NB, 128, 0, stream>>>(nom, den, shifts);

    // 6) fresh forward FFT2 of raw frames (src_freq), phase shift, inverse
    k_fft_rows_real<<<4096, 128, 0, stream>>>(fr, F, stats, 0);
    k_fft_cols<false><<<4096, 128, 0, stream>>>(F);
    k_phase<<<32768, 256, 0, stream>>>(F, shifts);
    k_fft_cols<true><<<4096, 128, 0, stream>>>(F);
    k_fft_rows_cplx<true><<<4096, 128, 0, stream>>>(F);

    // 7) transposed real output
    k_writeout<<<32768, 256, 0, stream>>>(F, out);
}